// GraphSAGE_83794811945493
// MI455X (gfx1250) — compile-verified
//
#include <hip/hip_runtime.h>
#include <hip/hip_bf16.h>

typedef __attribute__((ext_vector_type(2))) float v2f;
typedef __attribute__((ext_vector_type(8))) float v8f;

// ---------------- utility kernels ----------------

__global__ void zero_f32v4(float4* __restrict__ p, long long n4) {
    long long i = (long long)blockIdx.x * blockDim.x + threadIdx.x;
    if (i < n4) p[i] = make_float4(0.f, 0.f, 0.f, 0.f);
}

__global__ void zero_i32(int* __restrict__ p, int n) {
    int i = blockIdx.x * blockDim.x + threadIdx.x;
    if (i < n) p[i] = 0;
}

__global__ void sage_count(const int* __restrict__ dst, int* __restrict__ cnt, int nEdges) {
    int e = blockIdx.x * blockDim.x + threadIdx.x;
    if (e < nEdges) atomicAdd(&cnt[dst[e]], 1);
}

__global__ void sage_invcnt(const int* __restrict__ cnt, float* __restrict__ invc, int n) {
    int i = blockIdx.x * blockDim.x + threadIdx.x;
    if (i < n) {
        int c = cnt[i];
        invc[i] = 1.0f / (float)(c > 1 ? c : 1);
    }
}

__device__ __forceinline__ void atomAddF32(float* p, float v) {
    // hardware global_atomic_add_f32 (no return)
    unsafeAtomicAdd(p, v);
}

// scatter-add of feat[src] into aggr[dst]; 4 threads per edge, 16 channels each
__global__ void sage_scatter(const float* __restrict__ feat,
                             const int* __restrict__ src,
                             const int* __restrict__ dst,
                             float* __restrict__ aggr,
                             int nEdges) {
    long long t = (long long)blockIdx.x * blockDim.x + threadIdx.x;
    long long e = t >> 2;
    if (e >= nEdges) return;
    int part = ((int)t & 3) * 16;
    int s = src[e], d = dst[e];
    const float4* fp = (const float4*)(feat + (long long)s * 64 + part);
    float* ap = aggr + (long long)d * 64 + part;
#pragma unroll
    for (int i = 0; i < 4; ++i) {
        float4 v = fp[i];
        atomAddF32(ap + i * 4 + 0, v.x);
        atomAddF32(ap + i * 4 + 1, v.y);
        atomAddF32(ap + i * 4 + 2, v.z);
        atomAddF32(ap + i * 4 + 3, v.w);
    }
}

// ---------------- fused SAGE layer GEMM ----------------
// out[n, :] = act( (aggr[n,:]*invc[n]) @ Wl^T + b + x[n,:] @ Wr^T )
// One wave handles a 16-node row tile; f32 WMMA 16x16x4, K=64 (16 steps), N=64 (4 tiles).

__global__ __launch_bounds__(256) void sage_gemm_wmma(
    const float* __restrict__ xin,   // [N,64]
    const float* __restrict__ aggr,  // [N,64] (unnormalized sums)
    const float* __restrict__ invc,  // [N]
    const float* __restrict__ Wl,    // [64,64] row-major [out][in]
    const float* __restrict__ Wr,    // [64,64]
    const float* __restrict__ bias,  // [64]
    float* __restrict__ out,         // [N,64]
    int nNodes, int doRelu) {
    const int lane = threadIdx.x & 31;
    const int wave = threadIdx.x >> 5;
    const int tile = blockIdx.x * (blockDim.x >> 5) + wave;
    const int node0 = tile * 16;
    if (node0 + 16 > nNodes) return;  // wave-uniform guard (EXEC all-1s inside)

    const int mrow  = lane & 15;         // A-row / B-col / D-col index
    const int khalf = (lane >> 4) << 1;  // 0 or 2
    const int node  = node0 + mrow;
    const float inv = invc[node];

    v8f acc[4];
#pragma unroll
    for (int nt = 0; nt < 4; ++nt) {
        float bv = bias[nt * 16 + mrow];
#pragma unroll
        for (int r = 0; r < 8; ++r) acc[nt][r] = bv;
    }

    const float* aRow = aggr + (long long)node * 64 + khalf;
    const float* xRow = xin  + (long long)node * 64 + khalf;

    // aggr-path: A = aggr*inv, B = Wl^T
#pragma unroll
    for (int k = 0; k < 16; ++k) {
        v2f a = *(const v2f*)(aRow + k * 4);
        a *= inv;
#pragma unroll
        for (int nt = 0; nt < 4; ++nt) {
            v2f b = *(const v2f*)(Wl + (long long)(nt * 16 + mrow) * 64 + k * 4 + khalf);
            acc[nt] = __builtin_amdgcn_wmma_f32_16x16x4_f32(
                false, a, false, b, (short)0, acc[nt], false, false);
        }
    }
    // self-path: A = x, B = Wr^T
#pragma unroll
    for (int k = 0; k < 16; ++k) {
        v2f a = *(const v2f*)(xRow + k * 4);
#pragma unroll
        for (int nt = 0; nt < 4; ++nt) {
            v2f b = *(const v2f*)(Wr + (long long)(nt * 16 + mrow) * 64 + k * 4 + khalf);
            acc[nt] = __builtin_amdgcn_wmma_f32_16x16x4_f32(
                false, a, false, b, (short)0, acc[nt], false, false);
        }
    }

    // D layout: VGPR r, lane l -> row r + 8*(l>>4), col l&15
    const int rOff = (lane >> 4) * 8;
#pragma unroll
    for (int nt = 0; nt < 4; ++nt) {
#pragma unroll
        for (int r = 0; r < 8; ++r) {
            float v = acc[nt][r];
            if (doRelu) v = fmaxf(v, 0.0f);
            out[(long long)(node0 + rOff + r) * 64 + nt * 16 + mrow] = v;
        }
    }
}

// ---------------- host launch ----------------

extern "C" void kernel_launch(void* const* d_in, const int* in_sizes, int n_in,
                              void* d_out, int out_size, void* d_ws, size_t ws_size,
                              hipStream_t stream) {
    const float* x   = (const float*)d_in[0];
    const int*   ei  = (const int*)d_in[1];
    const float* W1l = (const float*)d_in[2];
    const float* b1  = (const float*)d_in[3];
    const float* W1r = (const float*)d_in[4];
    const float* W2l = (const float*)d_in[5];
    const float* b2  = (const float*)d_in[6];
    const float* W2r = (const float*)d_in[7];
    const float* W3l = (const float*)d_in[8];
    const float* b3  = (const float*)d_in[9];
    const float* W3r = (const float*)d_in[10];

    const int nNodes = in_sizes[0] / 64;
    const int nEdges = in_sizes[1] / 2;
    const int* src = ei;
    const int* dst = ei + nEdges;

    const long long feat_elems = (long long)nNodes * 64;
    float* aggr = (float*)d_ws;
    float* h1   = aggr + feat_elems;
    float* h2   = h1 + feat_elems;
    int*   cnt  = (int*)(h2 + feat_elems);
    float* invc = (float*)(cnt + nNodes);
    float* outp = (float*)d_out;

    const int B = 256;
    const long long zero4 = feat_elems / 4;
    const int gZero  = (int)((zero4 + B - 1) / B);
    const int gNode  = (nNodes + B - 1) / B;
    const int gEdge  = (nEdges + B - 1) / B;
    const int gScat  = (int)(((long long)nEdges * 4 + B - 1) / B);
    const int tiles  = nNodes / 16;
    const int gGemm  = (tiles + (B / 32) - 1) / (B / 32);

    // degree counts (shared by all layers)
    zero_i32<<<gNode, B, 0, stream>>>(cnt, nNodes);
    sage_count<<<gEdge, B, 0, stream>>>(dst, cnt, nEdges);
    sage_invcnt<<<gNode, B, 0, stream>>>(cnt, invc, nNodes);

    // layer 1
    zero_f32v4<<<gZero, B, 0, stream>>>((float4*)aggr, zero4);
    sage_scatter<<<gScat, B, 0, stream>>>(x, src, dst, aggr, nEdges);
    sage_gemm_wmma<<<gGemm, B, 0, stream>>>(x, aggr, invc, W1l, W1r, b1, h1, nNodes, 1);

    // layer 2
    zero_f32v4<<<gZero, B, 0, stream>>>((float4*)aggr, zero4);
    sage_scatter<<<gScat, B, 0, stream>>>(h1, src, dst, aggr, nEdges);
    sage_gemm_wmma<<<gGemm, B, 0, stream>>>(h1, aggr, invc, W2l, W2r, b2, h2, nNodes, 1);

    // layer 3 (no relu)
    zero_f32v4<<<gZero, B, 0, stream>>>((float4*)aggr, zero4);
    sage_scatter<<<gScat, B, 0, stream>>>(h2, src, dst, aggr, nEdges);
    sage_gemm_wmma<<<gGemm, B, 0, stream>>>(h2, aggr, invc, W3l, W3r, b3, outp, nNodes, 0);
}